// L0MLLP_32882269618837
// MI455X (gfx1250) — compile-verified
//
#include <hip/hip_runtime.h>

typedef float v2f __attribute__((ext_vector_type(2)));
typedef float v8f __attribute__((ext_vector_type(8)));

#define BATCH 64
#define IN_F  1024
#define OUT_F 1024
#define NPOW  8          // log(1-t) series order; t < 0.1 -> trunc err < 1.2e-10

// One block per 16x16 output tile (4 x 64 = 256 tiles).
// 8 waves per block, each wave accumulates a 128-wide K-chunk of the
// 8-term log-series GEMM on the fp32 WMMA pipe, then LDS-reduce + exp.
__global__ __launch_bounds__(256)
void l0conj_wmma_kernel(const float* __restrict__ x,        // [BATCH][IN_F]
                        const float* __restrict__ weights,  // [IN_F][OUT_F]
                        const float* __restrict__ qz,       // [IN_F]
                        float* __restrict__ out)            // [BATCH][OUT_F]
{
    __shared__ float zsh[IN_F];   // hard-concrete gate z[i]
    __shared__ float red[256];    // 16x16 partial log-sum reduction tile

    const int tid = threadIdx.x;

    // z = clamp(sigmoid(q) * (limit_b - limit_a) + limit_a, 0, 1)
    for (int k = tid; k < IN_F; k += 256) {
        float q  = qz[k];
        float pi = 1.0f / (1.0f + __expf(-q));
        float z  = pi * 1.2f - 0.1f;
        zsh[k]   = fminf(fmaxf(z, 0.0f), 1.0f);
    }
    red[tid] = 0.0f;
    __syncthreads();

    const int wave = tid >> 5;
    const int lane = tid & 31;
    const int half = lane >> 4;   // 0: lanes 0-15, 1: lanes 16-31
    const int l16  = lane & 15;

    const int tileId = blockIdx.x;   // 0..255
    const int tm = tileId >> 6;      // 0..3   (M tile: rows of x / out)
    const int tn = tileId & 63;      // 0..63  (N tile: cols of weights / out)

    const int row = tm * 16 + l16;   // A-matrix row (per-lane, both halves)
    const int col = tn * 16 + l16;   // B/C-matrix column

    v8f acc = {0.0f, 0.0f, 0.0f, 0.0f, 0.0f, 0.0f, 0.0f, 0.0f};
    const float invp[NPOW] = {1.0f, 0.5f, 1.0f/3.0f, 0.25f,
                              0.2f, 1.0f/6.0f, 1.0f/7.0f, 0.125f};

    // This wave's K-chunk of the inner dimension.
    const int i0 = wave * (IN_F / 8);
    const int i1 = i0 + (IN_F / 8);

    for (int i = i0; i < i1; i += 4) {
        // fp32 WMMA 16x16x4 A layout: lanes 0-15 hold K=i,i+1; lanes 16-31 K=i+2,i+3
        const int k = i + half * 2;

        float u0 = 1.0f - x[row * IN_F + k];
        float u1 = 1.0f - x[row * IN_F + k + 1];

        float w0 = zsh[k]     * weights[k       * OUT_F + col];
        float w1 = zsh[k + 1] * weights[(k + 1) * OUT_F + col];

        // Incremental powers: term_p = (u^p / p) * (w^p); acc += A_p x B_p
        float up0 = u0, up1 = u1, wp0 = w0, wp1 = w1;
#pragma unroll
        for (int p = 0; p < NPOW; ++p) {
            v2f a; a.x = up0 * invp[p]; a.y = up1 * invp[p];
            v2f b; b.x = wp0;           b.y = wp1;
            acc = __builtin_amdgcn_wmma_f32_16x16x4_f32(
                      /*neg_a=*/false, a, /*neg_b=*/false, b,
                      /*c_mod=*/(short)0, acc,
                      /*reuse_a=*/false, /*reuse_b=*/false);
            up0 *= u0; up1 *= u1;
            wp0 *= w0; wp1 *= w1;
        }
    }

    // Reduce the 8 waves' partial log-sums: C/D layout is
    // VGPR j -> M = j (lanes 0-15) or j+8 (lanes 16-31), N = lane%16.
#pragma unroll
    for (int j = 0; j < 8; ++j) {
        int mloc = j + half * 8;
        atomicAdd(&red[mloc * 16 + l16], acc[j]);   // ds_add_f32
    }
    __syncthreads();

    // out = exp(-sum_log_terms)
    {
        int mloc = tid >> 4;
        int nloc = tid & 15;
        out[(tm * 16 + mloc) * OUT_F + (tn * 16 + nloc)] = __expf(-red[tid]);
    }
}

extern "C" void kernel_launch(void* const* d_in, const int* in_sizes, int n_in,
                              void* d_out, int out_size, void* d_ws, size_t ws_size,
                              hipStream_t stream) {
    (void)in_sizes; (void)n_in; (void)out_size; (void)d_ws; (void)ws_size;
    const float* x       = (const float*)d_in[0];
    const float* weights = (const float*)d_in[1];
    const float* qz      = (const float*)d_in[2];
    float*       out     = (float*)d_out;

    // 256 tiles of 16x16 covering the 64x1024 output; 8 waves per block.
    l0conj_wmma_kernel<<<256, 256, 0, stream>>>(x, weights, qz, out);
}